// PaiNNInteraction_20761871909440
// MI455X (gfx1250) — compile-verified
//
#include <hip/hip_runtime.h>
#include <hip/hip_bf16.h>

#define F 128
#define TF 384           // 3*F
#define FP 132           // padded LDS row (132 % 64 == 4 -> conflict-free column reads)
#define EDGES_PER_WAVE 64

typedef float v2f __attribute__((ext_vector_type(2)));
typedef float v8f __attribute__((ext_vector_type(8)));

// ---------------------------------------------------------------------------
// Kernel 0: out = concat(q, mu)  (float4 coalesced)
// ---------------------------------------------------------------------------
__global__ __launch_bounds__(256) void painn_init_out(
    const float* __restrict__ q, const float* __restrict__ mu,
    float* __restrict__ out, int nq4, int ntot4)
{
  int i = blockIdx.x * blockDim.x + threadIdx.x;
  if (i >= ntot4) return;
  float4 v = (i < nq4) ? ((const float4*)q)[i] : ((const float4*)mu)[i - nq4];
  ((float4*)out)[i] = v;
}

// ---------------------------------------------------------------------------
// Kernel 1: X[a, 0:384] = silu(q[a]*W1 + b1) * W2 + b2   via f32 WMMA
// One block = 16 atoms. 8 waves: GEMM1 covers all 128 cols (8 tiles),
// GEMM2 covers 384 cols (3 tiles per wave).
// ---------------------------------------------------------------------------
__global__ __launch_bounds__(256) void painn_ctx_gemm(
    const float* __restrict__ q,   // [A,128]
    const float* __restrict__ W1,  // [128,128] (K-major rows)
    const float* __restrict__ b1,  // [128]
    const float* __restrict__ W2,  // [128,384]
    const float* __restrict__ b2,  // [384]
    float* __restrict__ X)         // [A,384]
{
  __shared__ float sA[16 * FP];
  __shared__ float sH[16 * FP];

  const int tid    = threadIdx.x;
  const int lane   = tid & 31;
  const int wave   = tid >> 5;          // 0..7
  const int m_base = blockIdx.x * 16;
  const int half   = lane >> 4;         // 0: K=k0,k0+1   1: K=k0+2,k0+3
  const int l15    = lane & 15;

  // ---- stage 16x128 q tile into LDS (padded rows) ----
  {
    const int row = tid >> 4;           // 0..15
    const int seg = tid & 15;           // 0..15 -> 8 floats each
    const float4* src = (const float4*)(q + (size_t)(m_base + row) * F) + seg * 2;
    float4* dst = (float4*)(sA + row * FP) + seg * 2;
    dst[0] = src[0];
    dst[1] = src[1];
  }
  __syncthreads();

  // ---- GEMM1: h = silu(qTile * W1 + b1), wave handles n_base = 16*wave ----
  {
    const int n_base = wave * 16;
    v8f c = {};
    for (int k0 = 0; k0 < F; k0 += 4) {
      const int ka = k0 + 2 * half;
      v2f a, b;
      a.x = sA[l15 * FP + ka];
      a.y = sA[l15 * FP + ka + 1];
      b.x = W1[(size_t)(ka    ) * F + n_base + l15];
      b.y = W1[(size_t)(ka + 1) * F + n_base + l15];
      c = __builtin_amdgcn_wmma_f32_16x16x4_f32(false, a, false, b,
                                                (short)0, c, false, false);
    }
    const float bias = b1[n_base + l15];
#pragma unroll
    for (int r = 0; r < 8; ++r) {
      float z = c[r] + bias;
      float s = z / (1.0f + __expf(-z));       // silu
      sH[(r + 8 * half) * FP + n_base + l15] = s;
    }
  }
  __syncthreads();

  // ---- GEMM2: X = h * W2 + b2, wave handles 3 of 24 N-tiles ----
  for (int t = 0; t < 3; ++t) {
    const int n_base = (wave + 8 * t) * 16;
    v8f c = {};
    for (int k0 = 0; k0 < F; k0 += 4) {
      const int ka = k0 + 2 * half;
      v2f a, b;
      a.x = sH[l15 * FP + ka];
      a.y = sH[l15 * FP + ka + 1];
      b.x = W2[(size_t)(ka    ) * TF + n_base + l15];
      b.y = W2[(size_t)(ka + 1) * TF + n_base + l15];
      c = __builtin_amdgcn_wmma_f32_16x16x4_f32(false, a, false, b,
                                                (short)0, c, false, false);
    }
    const float bias = b2[n_base + l15];
#pragma unroll
    for (int r = 0; r < 8; ++r) {
      const int m = m_base + r + 8 * half;
      X[(size_t)m * TF + n_base + l15] = c[r] + bias;
    }
  }
}

// ---------------------------------------------------------------------------
// Kernel 2: edge streamer. One wave per 64 contiguous edges; lane owns
// features [4*lane, 4*lane+4). Register-accumulate per sorted segment,
// flush with native f32 atomics on segment change.
// ---------------------------------------------------------------------------
__device__ __forceinline__ void fma4(float4& a, const float4& x, const float4& y) {
  a.x = fmaf(x.x, y.x, a.x); a.y = fmaf(x.y, y.y, a.y);
  a.z = fmaf(x.z, y.z, a.z); a.w = fmaf(x.w, y.w, a.w);
}
__device__ __forceinline__ void fma4s(float4& a, const float4& x, float s) {
  a.x = fmaf(x.x, s, a.x); a.y = fmaf(x.y, s, a.y);
  a.z = fmaf(x.z, s, a.z); a.w = fmaf(x.w, s, a.w);
}
__device__ __forceinline__ float4 mul4(const float4& x, const float4& y) {
  float4 r; r.x = x.x*y.x; r.y = x.y*y.y; r.z = x.z*y.z; r.w = x.w*y.w; return r;
}

__global__ __launch_bounds__(256) void painn_edge(
    const float* __restrict__ X,       // [A,384]
    const float* __restrict__ mu,      // [A,3,128] == [A,384]
    const float* __restrict__ Wij,     // [E,384]
    const float* __restrict__ dir_ij,  // [E,3]
    const int*   __restrict__ idx_i,
    const int*   __restrict__ idx_j,
    float* __restrict__ out_q,         // [A,128]
    float* __restrict__ out_mu,        // [A,3,128]
    int n_edges)
{
  const int lane = threadIdx.x & 31;
  const int wave = (blockIdx.x * blockDim.x + threadIdx.x) >> 5;
  const int e0 = wave * EDGES_PER_WAVE;
  if (e0 >= n_edges) return;
  const int e1 = min(e0 + EDGES_PER_WAVE, n_edges);

  float4 accq = {0.f, 0.f, 0.f, 0.f};
  float4 acc0 = accq, acc1 = accq, acc2 = accq;
  int cur_i = __builtin_amdgcn_readfirstlane(idx_i[e0]);

  auto flush = [&]() {
    float* pq = out_q + (size_t)cur_i * F + 4 * lane;
    unsafeAtomicAdd(pq + 0, accq.x); unsafeAtomicAdd(pq + 1, accq.y);
    unsafeAtomicAdd(pq + 2, accq.z); unsafeAtomicAdd(pq + 3, accq.w);
    float* pm = out_mu + (size_t)cur_i * TF + 4 * lane;
    unsafeAtomicAdd(pm + 0,       acc0.x); unsafeAtomicAdd(pm + 1,       acc0.y);
    unsafeAtomicAdd(pm + 2,       acc0.z); unsafeAtomicAdd(pm + 3,       acc0.w);
    unsafeAtomicAdd(pm + F + 0,   acc1.x); unsafeAtomicAdd(pm + F + 1,   acc1.y);
    unsafeAtomicAdd(pm + F + 2,   acc1.z); unsafeAtomicAdd(pm + F + 3,   acc1.w);
    unsafeAtomicAdd(pm + 2*F + 0, acc2.x); unsafeAtomicAdd(pm + 2*F + 1, acc2.y);
    unsafeAtomicAdd(pm + 2*F + 2, acc2.z); unsafeAtomicAdd(pm + 2*F + 3, acc2.w);
    accq = make_float4(0.f, 0.f, 0.f, 0.f);
    acc0 = accq; acc1 = accq; acc2 = accq;
  };

  for (int e = e0; e < e1; ++e) {
    const int i = __builtin_amdgcn_readfirstlane(idx_i[e]);
    const int j = __builtin_amdgcn_readfirstlane(idx_j[e]);
    if (i != cur_i) { flush(); cur_i = i; }   // wave-uniform branch (idx_i sorted)

    const float4* wp = (const float4*)(Wij + (size_t)e * TF);
    if (e + 1 < n_edges) {                    // warm next edge's Wij stream
      __builtin_prefetch((const void*)(Wij + (size_t)(e + 1) * TF + lane * 12), 0, 1);
    }
    const float4* xp = (const float4*)(X  + (size_t)j * TF);
    const float4* mp = (const float4*)(mu + (size_t)j * TF);

    float4 wq = wp[lane],      wr = wp[32 + lane], wm = wp[64 + lane];
    float4 xq = xp[lane],      xr = xp[32 + lane], xm = xp[64 + lane];
    float4 m0 = mp[lane],      m1 = mp[32 + lane], m2 = mp[64 + lane];
    const float dx = dir_ij[3 * (size_t)e + 0];
    const float dy = dir_ij[3 * (size_t)e + 1];
    const float dz = dir_ij[3 * (size_t)e + 2];

    fma4(accq, wq, xq);                       // dq        += Wq  * xj_q
    float4 dr = mul4(wr, xr);                 // dmuR       = WmR * xj_mR
    float4 dm = mul4(wm, xm);                 // dmumu      = Wmm * xj_mm
    fma4s(acc0, dr, dx); fma4(acc0, dm, m0);  // dmu[d]    += dr*dir_d + dm*mu_j[d]
    fma4s(acc1, dr, dy); fma4(acc1, dm, m1);
    fma4s(acc2, dr, dz); fma4(acc2, dm, m2);
  }
  flush();
}

// ---------------------------------------------------------------------------
extern "C" void kernel_launch(void* const* d_in, const int* in_sizes, int n_in,
                              void* d_out, int out_size, void* d_ws, size_t ws_size,
                              hipStream_t stream) {
  const float* q     = (const float*)d_in[0];
  const float* mu    = (const float*)d_in[1];
  const float* Wij   = (const float*)d_in[2];
  const float* dir   = (const float*)d_in[3];
  const int*   idx_i = (const int*)d_in[4];
  const int*   idx_j = (const int*)d_in[5];
  // d_in[6] = n_atoms (device scalar) -- derive on host instead
  const float* W1    = (const float*)d_in[7];
  const float* b1    = (const float*)d_in[8];
  const float* W2    = (const float*)d_in[9];
  const float* b2    = (const float*)d_in[10];

  const int n_atoms = in_sizes[0] / F;       // 20000
  const int n_edges = in_sizes[4];           // 640000

  float* X      = (float*)d_ws;              // [n_atoms, 384] scratch (30.7 MB)
  float* out_q  = (float*)d_out;             // [n_atoms, 128]
  float* out_mu = out_q + (size_t)n_atoms * F;

  // 0) out = concat(q, mu)
  const int nq4   = n_atoms * F / 4;
  const int ntot4 = nq4 + n_atoms * TF / 4;
  painn_init_out<<<(ntot4 + 255) / 256, 256, 0, stream>>>(q, mu, (float*)d_out, nq4, ntot4);

  // 1) X = (silu(q*W1+b1))*W2 + b2   (WMMA f32; n_atoms divisible by 16)
  painn_ctx_gemm<<<n_atoms / 16, 256, 0, stream>>>(q, W1, b1, W2, b2, X);

  // 2) edge stream + segmented atomic scatter
  const int waves  = (n_edges + EDGES_PER_WAVE - 1) / EDGES_PER_WAVE;
  const int blocks = (waves * 32 + 255) / 256;
  painn_edge<<<blocks, 256, 0, stream>>>(X, mu, Wij, dir, idx_i, idx_j,
                                         out_q, out_mu, n_edges);
}